// ClusteringLayer_4114578670341
// MI455X (gfx1250) — compile-verified
//
#include <hip/hip_runtime.h>
#include <hip/hip_bf16.h>

typedef __attribute__((ext_vector_type(2))) float v2f;
typedef __attribute__((ext_vector_type(8))) float v8f;

#define KNB   32      // max neighbors
#define CAP   256     // per-row candidate cap (mean in-ball count ~69)
#define TILE  16
#define WAVES 8       // 256 threads / wave32
#define SENTINEL 1e30f

__global__ __launch_bounds__(256)
void radius_graph_wmma_kernel(const float* __restrict__ pos,
                              float* __restrict__ out_src,
                              float* __restrict__ out_dst,
                              float* __restrict__ out_mask,
                              int N, float r2)
{
    __shared__ float sqRow[TILE];
    __shared__ int   scnt[TILE];
    __shared__ float cd2[TILE][CAP];
    __shared__ int   cidx[TILE][CAP];

    const int tid  = threadIdx.x;
    const int lane = tid & 31;
    const int wave = tid >> 5;
    const int rowBase = blockIdx.x * TILE;

    if (tid < TILE) {
        int r = rowBase + tid;
        float x = pos[3*r+0], y = pos[3*r+1], z = pos[3*r+2];
        sqRow[tid] = x*x + y*y + z*z;
        scnt[tid]  = 0;
    }
    __syncthreads();

    // ---------- Phase 1: WMMA distance tiles + candidate gather ----------
    const int lm = lane & 15;   // M (for A) / N (for B) within 16-tile
    const int hi = lane >> 4;   // lane half selects K-pair {0,1} vs {2,3}

    // A matrix 16x4 (f32): row m = (x, y, z, 1)
    const int arow = rowBase + lm;
    const float ax = pos[3*arow+0], ay = pos[3*arow+1], az = pos[3*arow+2];
    v2f a;
    a[0] = hi ? az   : ax;
    a[1] = hi ? 1.0f : ay;

    // C seed: c[v] = |p_row|^2, row = v + 8*hi
    v8f cinit;
    #pragma unroll
    for (int v = 0; v < 8; ++v) cinit[v] = sqRow[v + 8*hi];

    const int numTiles = N / TILE;
    for (int ct = wave; ct < numTiles; ct += WAVES) {
        // B matrix 4x16 (f32): col n = (-2x, -2y, -2z, |p_n|^2)
        const int col = ct * TILE + lm;
        const float bx = pos[3*col+0], by = pos[3*col+1], bz = pos[3*col+2];
        const float sqc = bx*bx + by*by + bz*bz;
        v2f b;
        b[0] = hi ? (-2.0f*bz) : (-2.0f*bx);
        b[1] = hi ? sqc        : (-2.0f*by);

        // D(m,n) = -2 p_m.p_n + |p_n|^2 + |p_m|^2 = squared distance
        v8f d = __builtin_amdgcn_wmma_f32_16x16x4_f32(
            /*neg_a=*/false, a, /*neg_b=*/false, b,
            /*c_mod=*/(short)0, cinit, /*reuse_a=*/false, /*reuse_b=*/false);

        #pragma unroll
        for (int v = 0; v < 8; ++v) {
            const int m  = v + 8*hi;
            const int rg = rowBase + m;
            const float d2 = d[v];
            if (d2 < r2 && col != rg) {
                int slot = atomicAdd(&scnt[m], 1);
                if (slot < CAP) { cd2[m][slot] = d2; cidx[m][slot] = col; }
            }
        }
    }
    __syncthreads();

    // ---------- Phase 2: per-row nearest-32 selection (register resident) ----------
    for (int rr = 0; rr < 2; ++rr) {
        const int m    = wave * 2 + rr;       // 8 waves x 2 rows = 16 rows
        const int rowG = rowBase + m;
        int cnt = scnt[m]; if (cnt > CAP) cnt = CAP;

        // Strided slice of the candidate list into registers
        float ld2[8]; int lix[8];
        #pragma unroll
        for (int t = 0; t < 8; ++t) {
            const int s = lane + 32*t;
            const bool ok = s < cnt;
            ld2[t] = ok ? cd2[m][s] : SENTINEL;
            lix[t] = ok ? cidx[m][s] : 0x7fffffff;
        }

        for (int k = 0; k < KNB; ++k) {
            // local min (ascending d2, index tiebreak to mimic top_k ordering)
            float bd = SENTINEL; int bi = 0x7fffffff; int bt = -1;
            #pragma unroll
            for (int t = 0; t < 8; ++t) {
                if (ld2[t] < bd || (ld2[t] == bd && lix[t] < bi)) {
                    bd = ld2[t]; bi = lix[t]; bt = t;
                }
            }
            // wave32 butterfly reduction, tracking the winning lane
            int bl = lane;
            #pragma unroll
            for (int off = 16; off > 0; off >>= 1) {
                const float od = __shfl_xor(bd, off, 32);
                const int   oi = __shfl_xor(bi, off, 32);
                const int   ol = __shfl_xor(bl, off, 32);
                if (od < bd || (od == bd && oi < bi)) { bd = od; bi = oi; bl = ol; }
            }
            const bool found = (bd < SENTINEL);
            if (lane == 0) {
                const int o = rowG * KNB + k;
                out_src[o]  = found ? (float)bi   : -1.0f;
                out_dst[o]  = found ? (float)rowG : -1.0f;
                out_mask[o] = found ? 1.0f        : 0.0f;
            }
            // remove the winner from its owner lane's register slice
            if (found && lane == bl) {
                #pragma unroll
                for (int t = 0; t < 8; ++t) if (bt == t) ld2[t] = SENTINEL;
            }
        }
    }
}

extern "C" void kernel_launch(void* const* d_in, const int* in_sizes, int n_in,
                              void* d_out, int out_size, void* d_ws, size_t ws_size,
                              hipStream_t stream)
{
    (void)n_in; (void)out_size; (void)d_ws; (void)ws_size;

    const float* feature = (const float*)d_in[0];
    const float* pos     = (const float*)d_in[1];

    const int N = in_sizes[1] / 3;            // 16384
    const size_t featN = (size_t)in_sizes[0]; // N*D
    const size_t posN  = (size_t)N * 3;

    float* out   = (float*)d_out;
    float* oFeat = out;
    float* oPos  = oFeat + featN;
    float* oSrc  = oPos  + posN;
    float* oDst  = oSrc  + (size_t)N * KNB;
    float* oMask = oDst  + (size_t)N * KNB;

    // passthrough outputs (graph-capture-safe async D2D copies)
    hipMemcpyAsync(oFeat, feature, featN * sizeof(float), hipMemcpyDeviceToDevice, stream);
    hipMemcpyAsync(oPos,  pos,     posN  * sizeof(float), hipMemcpyDeviceToDevice, stream);

    const float r  = 0.1f * 0.999f;
    radius_graph_wmma_kernel<<<N / TILE, 256, 0, stream>>>(pos, oSrc, oDst, oMask, N, r * r);
}